// ContrastiveLoss_61280593380002
// MI455X (gfx1250) — compile-verified
//
#include <hip/hip_runtime.h>

// SimCLR-style contrastive loss, algebraically reduced:
//   loss = (1/(2B*T)) * sum_k [ (a.a)/na^2 + (b.b)/nb^2 - 2*(a.b)/(na*nb) ]
// where na = max(||a||,eps), nb = max(||b||,eps).
// Per-pair dot products computed with V_WMMA_F32_16X16X4_F32 (fp32 WMMA).

typedef __attribute__((ext_vector_type(2))) float v2f;
typedef __attribute__((ext_vector_type(4))) float v4f;
typedef __attribute__((ext_vector_type(8))) float v8f;

#define BB   4096
#define DD   512
#define TEMP 0.5f
#define EPSN 1e-12f

__device__ __forceinline__ float pick8(v8f c, int i) {
  // runtime select of one element out of a v8f accumulator (unrolled cndmask chain)
  float r = c[0];
  r = (i == 1) ? c[1] : r;
  r = (i == 2) ? c[2] : r;
  r = (i == 3) ? c[3] : r;
  r = (i == 4) ? c[4] : r;
  r = (i == 5) ? c[5] : r;
  r = (i == 6) ? c[6] : r;
  r = (i == 7) ? c[7] : r;
  return r;
}

// One block = 16 pairs. 4 waves split the K=512 dimension (128 cols each).
__global__ void __launch_bounds__(128)
simclr_pair_kernel(const float* __restrict__ ei, const float* __restrict__ ej,
                   float* __restrict__ partial) {
  const int tid  = threadIdx.x;
  const int lane = tid & 31;
  const int wave = tid >> 5;            // 0..3 : K-split
  const int r0   = blockIdx.x * 16;     // first pair row of this block
  const int m    = lane & 15;           // A-row index == B-col index (symmetric case)
  const int h    = lane >> 4;           // lane-half selects K sub-block

  const float* rowI = ei + (size_t)(r0 + m) * DD;
  const float* rowJ = ej + (size_t)(r0 + m) * DD;

  v8f cij = {};  // diag -> a.b
  v8f cii = {};  // diag -> a.a
  v8f cjj = {};  // diag -> b.b

  const int kBase = wave * (DD / 4);    // 128 columns per wave

  // 16 outer iterations, 8 columns each: one float4 load per array per lane,
  // consumed by two K=4 WMMA steps (K-slot -> column mapping is a permutation,
  // identical on the A and B operands, so the dot products are exact).
  #pragma unroll 4
  for (int t = 0; t < (DD / 4) / 8; ++t) {
    const int c0 = kBase + t * 8 + h * 4;
    v4f a = *(const v4f*)(rowI + c0);
    v4f b = *(const v4f*)(rowJ + c0);
    v2f a1 = {a.x, a.y};
    v2f a2 = {a.z, a.w};
    v2f b1 = {b.x, b.y};
    v2f b2 = {b.z, b.w};
    cij = __builtin_amdgcn_wmma_f32_16x16x4_f32(false, a1, false, b1, (short)0, cij, false, false);
    cii = __builtin_amdgcn_wmma_f32_16x16x4_f32(false, a1, false, a1, (short)0, cii, false, false);
    cjj = __builtin_amdgcn_wmma_f32_16x16x4_f32(false, b1, false, b1, (short)0, cjj, false, false);
    cij = __builtin_amdgcn_wmma_f32_16x16x4_f32(false, a2, false, b2, (short)0, cij, false, false);
    cii = __builtin_amdgcn_wmma_f32_16x16x4_f32(false, a2, false, a2, (short)0, cii, false, false);
    cjj = __builtin_amdgcn_wmma_f32_16x16x4_f32(false, b2, false, b2, (short)0, cjj, false, false);
  }

  // Extract diagonal: m<8 lives in lane m, VGPR m; m>=8 in lane m+16, VGPR m-8.
  __shared__ float sAB[4][16];
  __shared__ float sAA[4][16];
  __shared__ float sBB[4][16];

  const bool ownLow  = (lane < 8);
  const bool ownHigh = (lane >= 24);
  if (ownLow || ownHigh) {
    const int vidx = ownLow ? lane : (lane - 24);
    const int dm   = ownLow ? lane : (lane - 16);
    sAB[wave][dm] = pick8(cij, vidx);
    sAA[wave][dm] = pick8(cii, vidx);
    sBB[wave][dm] = pick8(cjj, vidx);
  }
  __syncthreads();

  if (wave == 0) {
    const int dm = lane & 15;
    float ab = 0.f, aa = 0.f, bb = 0.f;
    #pragma unroll
    for (int w = 0; w < 4; ++w) {
      ab += sAB[w][dm];
      aa += sAA[w][dm];
      bb += sBB[w][dm];
    }
    const float na  = fmaxf(sqrtf(aa), EPSN);
    const float nb  = fmaxf(sqrtf(bb), EPSN);
    const float dii = aa / (na * na);           // diag(sim) for row k      (~1)
    const float djj = bb / (nb * nb);           // diag(sim) for row k+B    (~1)
    const float p   = ab / (na * nb);           // positives (appears twice)
    float term = (lane < 16) ? (dii + djj - 2.0f * p) : 0.0f;
    #pragma unroll
    for (int off = 16; off > 0; off >>= 1)
      term += __shfl_xor(term, off, 32);
    if (lane == 0) partial[blockIdx.x] = term;
  }
}

// Deterministic final reduction of the 256 block partials.
__global__ void __launch_bounds__(256)
simclr_reduce_kernel(const float* __restrict__ partial, float* __restrict__ out) {
  __shared__ float red[8];
  const int tid  = threadIdx.x;
  const int lane = tid & 31;
  const int wid  = tid >> 5;
  float v = partial[tid];
  #pragma unroll
  for (int off = 16; off > 0; off >>= 1)
    v += __shfl_xor(v, off, 32);
  if (lane == 0) red[wid] = v;
  __syncthreads();
  if (wid == 0) {
    float s = (lane < 8) ? red[lane] : 0.0f;
    #pragma unroll
    for (int off = 4; off > 0; off >>= 1)
      s += __shfl_xor(s, off, 32);
    if (lane == 0)
      out[0] = s * (1.0f / (2.0f * (float)BB * TEMP));   // = s / 4096
  }
}

extern "C" void kernel_launch(void* const* d_in, const int* in_sizes, int n_in,
                              void* d_out, int out_size, void* d_ws, size_t ws_size,
                              hipStream_t stream) {
  (void)in_sizes; (void)n_in; (void)out_size; (void)ws_size;
  const float* ei = (const float*)d_in[0];
  const float* ej = (const float*)d_in[1];
  float* out      = (float*)d_out;
  float* partials = (float*)d_ws;      // 256 floats of scratch

  simclr_pair_kernel<<<BB / 16, 128, 0, stream>>>(ei, ej, partials);
  simclr_reduce_kernel<<<1, 256, 0, stream>>>(partials, out);
}